// SparsePropagation_16673063043308
// MI455X (gfx1250) — compile-verified
//
#include <hip/hip_runtime.h>

typedef __attribute__((ext_vector_type(2))) float v2f;
typedef __attribute__((ext_vector_type(4))) float f4;
typedef __attribute__((ext_vector_type(8))) float v8f;

#define WIN       128
#define NSEQ      4096
#define DDIM      256
#define WROWS     144          // window rows per i-tile: 128 + 16
#define VW_STRIDE 260          // LDS val row stride (floats), padded vs 64 banks
#define E_STRIDE  148          // LDS edges row stride (floats), padded

__global__ __launch_bounds__(128)
void sparse_prop_kernel(const float* __restrict__ val,
                        const float* __restrict__ state,
                        float* __restrict__ dstate,
                        float* __restrict__ dval) {
  extern __shared__ float smem[];
  float* val_w = smem;                         // [144][VW_STRIDE]
  float* edges = smem + WROWS * VW_STRIDE;     // [16][E_STRIDE]
  float* st_s  = edges + 16 * E_STRIDE;        // [144]

  const int tid  = threadIdx.x;
  const int lane = tid & 31;
  const int wave = tid >> 5;
  const int m    = lane & 15;                  // M index (A) / N index (B, C/D)
  const int h    = lane >> 4;                  // half-wave select

  const int blk = blockIdx.x;
  const int b   = blk >> 8;                    // 256 i-tiles per batch
  const int i0  = (blk & 255) << 4;
  const int j0  = i0 - WIN;                    // window start (may be < 0)

  const float* valb = val + (size_t)b * NSEQ * DDIM;

  // ---- Phase 0: stage window rows [j0, i0+16) of val into LDS (clamped) ----
  for (int t = tid; t < WROWS * (DDIM / 4); t += 128) {   // 72 iters/thread
    int row = t >> 6;                          // 64 float4 per row
    int c4  = t & 63;
    int jr  = j0 + row; if (jr < 0) jr = 0;    // clamped rows get masked later
    f4 v4 = *(const f4*)(valb + (size_t)jr * DDIM + c4 * 4);
    *(f4*)(val_w + row * VW_STRIDE + c4 * 4) = v4;
  }
  for (int t = tid; t < WROWS; t += 128) {
    int jr = j0 + t; if (jr < 0) jr = 0;
    st_s[t] = state[(size_t)b * NSEQ + jr];
  }
  __syncthreads();

  // ---- Phase 1: scores -> softsign -> causal-window mask -> edges in LDS ----
  // A[m][k] = val_w[128+m][k] (query rows), B[k][n] = val_w[jt*16+n][k].
  // Both operands share the same per-lane pattern: lane 16h+x loads the
  // contiguous f32 pair at column k0+2h of its row.
  for (int jt = wave; jt < 9; jt += 4) {
    v8f acc = {};
    const float* arow = val_w + (WIN + m) * VW_STRIDE;       // query row m
    const float* brow = val_w + (jt * 16 + m) * VW_STRIDE;   // source row m
    for (int k0 = 0; k0 < DDIM; k0 += 4) {
      v2f a  = *(const v2f*)(arow + k0 + 2 * h);
      v2f bm = *(const v2f*)(brow + k0 + 2 * h);
      acc = __builtin_amdgcn_wmma_f32_16x16x4_f32(false, a, false, bm,
                                                  (short)0, acc, false, false);
    }
    const int c = jt * 16 + m;                 // window-local column (N index)
    #pragma unroll
    for (int v = 0; v < 8; ++v) {
      int   r = v + 8 * h;                     // i-tile row (M index)
      float s = acc[v] * 0.0625f;              // 1/sqrt(256)
      // softsign: denominator >= 1, so a single v_rcp_f32 is safe (~1 ulp);
      // avoids the IEEE div_scale/Newton chain the plain '/' emitted.
      float e = s * __builtin_amdgcn_rcpf(1.0f + fabsf(s));
      bool ok = (c >= r) && (c <= r + WIN) && ((j0 + c) >= 0);
      edges[r * E_STRIDE + c] = ok ? e : 0.0f;
    }
  }
  __syncthreads();

  // ---- Phase 2a: delta_val = edges(16x144) @ val_w(144x256); waves split D --
  float* dvb = dval + ((size_t)b * NSEQ + i0) * DDIM;
  for (int dt = 0; dt < 4; ++dt) {
    int d0 = wave * 64 + dt * 16;
    v8f acc = {};
    for (int c0 = 0; c0 < WROWS; c0 += 4) {
      v2f a = *(const v2f*)(edges + m * E_STRIDE + c0 + 2 * h);
      v2f bm;
      bm.x = val_w[(c0 + 2 * h    ) * VW_STRIDE + d0 + m];
      bm.y = val_w[(c0 + 2 * h + 1) * VW_STRIDE + d0 + m];
      acc = __builtin_amdgcn_wmma_f32_16x16x4_f32(false, a, false, bm,
                                                  (short)0, acc, false, false);
    }
    #pragma unroll
    for (int v = 0; v < 8; ++v) {
      int r = v + 8 * h;
      dvb[(size_t)r * DDIM + d0 + m] = acc[v];
    }
  }

  // ---- Phase 2b: delta_state = edges(16x144) @ state(144); tiny matvec ----
  if (wave == 0 && lane < 16) {
    float s = 0.0f;
    for (int c = 0; c < WROWS; ++c) s += edges[lane * E_STRIDE + c] * st_s[c];
    dstate[(size_t)b * NSEQ + i0 + lane] = s;
  }
}

extern "C" void kernel_launch(void* const* d_in, const int* in_sizes, int n_in,
                              void* d_out, int out_size, void* d_ws, size_t ws_size,
                              hipStream_t stream) {
  const float* val   = (const float*)d_in[0];   // [4,4096,256] f32
  const float* state = (const float*)d_in[1];   // [4,4096]     f32
  float* dstate = (float*)d_out;                 // first output, B*N floats
  float* dval   = (float*)d_out + 4 * NSEQ;      // second output, B*N*D floats

  size_t smem_bytes =
      (size_t)(WROWS * VW_STRIDE + 16 * E_STRIDE + WROWS) * sizeof(float);

  hipLaunchKernelGGL(sparse_prop_kernel, dim3(1024), dim3(128), smem_bytes,
                     stream, val, state, dstate, dval);
}